// PrototypeRoutingModule_83305185673838
// MI455X (gfx1250) — compile-verified
//
#include <hip/hip_runtime.h>

// ---------------------------------------------------------------------------
// PrototypeRoutingModule for MI455X (gfx1250, wave32, WMMA)
// Shapes (from reference setup): B=4, D=64, N=2048, T=64, K=32, N_obs=1024
// ---------------------------------------------------------------------------

typedef __attribute__((ext_vector_type(2))) float v2f;
typedef __attribute__((ext_vector_type(8))) float v8f;

#define BB 4
#define DD 64
#define NN 2048
#define TT 64
#define KK 32
#define JJ (DD * TT)      // 4096 fused (d,t) column dimension
#define EPS_F 1e-8f

// ---------------------------------------------------------------------------
// Flags: obs_node boolean per node
// ---------------------------------------------------------------------------
__global__ void flags_zero_kernel(int* flags) {
    int i = blockIdx.x * blockDim.x + threadIdx.x;
    if (i < NN) flags[i] = 0;
}

__global__ void flags_set_kernel(const int* __restrict__ idx, int* flags, int n_obs) {
    int i = blockIdx.x * blockDim.x + threadIdx.x;
    if (i < n_obs) flags[idx[i]] = 1;
}

// ---------------------------------------------------------------------------
// K1: masked mean summary  s[b,n,d] = sum_t h[b,d,n,t]*m[b,n,t] / max(sum_t m, 1)
// grid = B*N blocks, 64 threads (thread = d).  mask tile staged in LDS.
// ---------------------------------------------------------------------------
__global__ __launch_bounds__(64) void summary_kernel(const float* __restrict__ h,
                                                     const float* __restrict__ mask,
                                                     float* __restrict__ s) {
    __shared__ float mlds[TT];
    int bid = blockIdx.x;
    int b = bid >> 11;            // / NN
    int n = bid & (NN - 1);
    int d = threadIdx.x;

    mlds[d] = mask[(b * NN + n) * TT + d];
    __syncthreads();

    float msum = 0.f;
#pragma unroll
    for (int t = 0; t < TT; ++t) msum += mlds[t];
    msum = fmaxf(msum, 1.0f);

    const float* hp = h + (((size_t)(b * DD + d) * NN + n) * TT);
    float acc = 0.f;
#pragma unroll
    for (int t = 0; t < TT; t += 4) {
        float4 hv = *(const float4*)(hp + t);
        acc += hv.x * mlds[t] + hv.y * mlds[t + 1] + hv.z * mlds[t + 2] + hv.w * mlds[t + 3];
    }
    s[(b * NN + n) * DD + d] = acc / msum;
}

// ---------------------------------------------------------------------------
// K2: alpha[b,n,k] = softmax_k( cos(s[b,n,:], proto[k,:]) )
// 256 threads = 8 wave32; one wave per node, lane = k (K==32==wave32).
// ---------------------------------------------------------------------------
__global__ __launch_bounds__(256) void alpha_kernel(const float* __restrict__ s,
                                                    const float* __restrict__ protos,
                                                    float* __restrict__ alpha) {
    __shared__ float pl[KK * DD];
    __shared__ float pinv[KK];
    int tid = threadIdx.x;

    for (int i = tid; i < KK * DD; i += 256) pl[i] = protos[i];
    __syncthreads();
    if (tid < KK) {
        float ss = 0.f;
#pragma unroll
        for (int d = 0; d < DD; ++d) { float v = pl[tid * DD + d]; ss += v * v; }
        pinv[tid] = 1.0f / fmaxf(sqrtf(ss), 1e-12f);
    }
    __syncthreads();

    int wave = tid >> 5, lane = tid & 31;
    int node = blockIdx.x * 8 + wave;      // 0 .. B*N-1
    int b = node >> 11;
    int n = node & (NN - 1);

    const float* sp = s + (b * NN + n) * DD;
    float dot = 0.f, ss = 0.f;
#pragma unroll
    for (int d = 0; d < DD; ++d) {
        float sv = sp[d];                  // broadcast load (same addr all lanes)
        dot += sv * pl[lane * DD + d];
        ss += sv * sv;
    }
    float sim = dot * pinv[lane] / fmaxf(sqrtf(ss), 1e-12f);

    // wave32 softmax over k = lane
    float mx = sim;
#pragma unroll
    for (int o = 16; o > 0; o >>= 1) mx = fmaxf(mx, __shfl_xor(mx, o, 32));
    float e = __expf(sim - mx);
    float sum = e;
#pragma unroll
    for (int o = 16; o > 0; o >>= 1) sum += __shfl_xor(sum, o, 32);

    alpha[(b * NN + n) * KK + lane] = e / sum;
}

// ---------------------------------------------------------------------------
// K3: H_num[b,k,j] = sum_{n' in obs} alpha[b,idx[n'],k] * h[b, j, idx[n']]
// One wave per 16-wide j tile; two v8f accumulators (k 0-15, 16-31) share the
// B fragment so h_obs is streamed exactly once.  Reduction: n' step 4 via
// v_wmma_f32_16x16x4_f32.
// ---------------------------------------------------------------------------
__global__ __launch_bounds__(256) void hnum_kernel(const float* __restrict__ h,
                                                   const float* __restrict__ alpha,
                                                   const int* __restrict__ idx,
                                                   float* __restrict__ hnum,
                                                   int n_obs) {
    int wave = threadIdx.x >> 5;
    int lane = threadIdx.x & 31;
    int gw = blockIdx.x * 8 + wave;        // 0 .. B*(JJ/16)-1
    int b  = gw >> 8;                      // / 256
    int jt = gw & 255;
    int j0 = jt * 16;
    int d  = j0 >> 6;
    int t0 = j0 & (TT - 1);

    int m    = lane & 15;                  // A row / B col
    int half = lane >> 4;
    int c0   = half * 2;                   // reduction sub-index {0,2}

    const float* hrow = h + ((size_t)(b * DD + d) * NN) * TT + t0 + m;
    const float* arow = alpha + (size_t)b * NN * KK;

    v8f acc0 = {}; v8f acc1 = {};
    for (int np = 0; np < n_obs; np += 4) {
        int nx = idx[np + c0];
        int ny = idx[np + c0 + 1];
        const float* ax = arow + nx * KK;
        const float* ay = arow + ny * KK;
        v2f a0, a1, bf;
        a0.x = ax[m];        a0.y = ay[m];          // k rows 0..15
        a1.x = ax[m + 16];   a1.y = ay[m + 16];     // k rows 16..31
        bf.x = hrow[(size_t)nx * TT];
        bf.y = hrow[(size_t)ny * TT];
        acc0 = __builtin_amdgcn_wmma_f32_16x16x4_f32(false, a0, false, bf, (short)0, acc0, false, false);
        acc1 = __builtin_amdgcn_wmma_f32_16x16x4_f32(false, a1, false, bf, (short)0, acc1, false, false);
    }

#pragma unroll
    for (int v = 0; v < 8; ++v) {
        int k0 = v + half * 8;
        hnum[((size_t)(b * KK + k0)      * JJ) + j0 + m] = acc0[v];
        hnum[((size_t)(b * KK + k0 + 16) * JJ) + j0 + m] = acc1[v];
    }
}

// ---------------------------------------------------------------------------
// K4: rdenom[b,k] = 1 / max(sum_{n' in obs} alpha[b,idx[n'],k], EPS)
// ---------------------------------------------------------------------------
__global__ __launch_bounds__(128) void rdenom_kernel(const float* __restrict__ alpha,
                                                     const int* __restrict__ idx,
                                                     float* __restrict__ rdenom,
                                                     int n_obs) {
    int t = threadIdx.x;                   // 0..127 -> (b,k)
    int b = t >> 5;
    int k = t & 31;
    float sum = 0.f;
    for (int np = 0; np < n_obs; ++np)
        sum += alpha[((size_t)b * NN + idx[np]) * KK + k];
    rdenom[t] = 1.0f / fmaxf(sum, EPS_F);
}

// ---------------------------------------------------------------------------
// K5: out[b,d,n,t] = obs[n] ? h : sum_k alpha[b,n,k]*rdenom[b,k]*H_num[b,k,j]
// One wave per 16n x 16j tile; 8 x v_wmma_f32_16x16x4_f32 over k=32.
// Fully-observed tiles take a wave-uniform copy fast path.
// ---------------------------------------------------------------------------
__global__ __launch_bounds__(256) void impute_kernel(const float* __restrict__ h,
                                                     const float* __restrict__ alpha,
                                                     const float* __restrict__ hnum,
                                                     const float* __restrict__ rdenom,
                                                     const int* __restrict__ flags,
                                                     float* __restrict__ out) {
    int wave = threadIdx.x >> 5;
    int lane = threadIdx.x & 31;
    int gw = blockIdx.x * 8 + wave;        // 0 .. B*(N/16)*(JJ/16)-1
    int b   = gw >> 15;                    // / (128*256)
    int rem = gw & 32767;
    int nt  = rem >> 8;
    int jt  = rem & 255;
    int n0  = nt * 16;
    int j0  = jt * 16;
    int d   = j0 >> 6;
    int t0  = j0 & (TT - 1);

    int m    = lane & 15;
    int half = lane >> 4;
    int c0   = half * 2;

    // fast path: all 16 node-rows observed -> straight copy
    int myflag = flags[n0 + m];
    if (__all(myflag != 0)) {
#pragma unroll
        for (int v = 0; v < 8; ++v) {
            int n = n0 + v + half * 8;
            size_t oi = ((size_t)(b * DD + d) * NN + n) * TT + t0 + m;
            out[oi] = h[oi];
        }
        return;
    }

    const float* arow = alpha + ((size_t)b * NN + n0 + m) * KK;
    const float* hp   = hnum + (size_t)b * KK * JJ + j0 + m;
    const float* rd   = rdenom + b * KK;

    v8f acc = {};
#pragma unroll
    for (int ks = 0; ks < 8; ++ks) {
        int kk = ks * 4 + c0;
        v2f a, bf;
        a.x = arow[kk]     * rd[kk];
        a.y = arow[kk + 1] * rd[kk + 1];
        bf.x = hp[(size_t)kk * JJ];
        bf.y = hp[(size_t)(kk + 1) * JJ];
        acc = __builtin_amdgcn_wmma_f32_16x16x4_f32(false, a, false, bf, (short)0, acc, false, false);
    }

#pragma unroll
    for (int v = 0; v < 8; ++v) {
        int n = n0 + v + half * 8;
        size_t oi = ((size_t)(b * DD + d) * NN + n) * TT + t0 + m;
        float val = acc[v];
        if (flags[n]) val = h[oi];
        out[oi] = val;
    }
}

// ---------------------------------------------------------------------------
// Launch
// ---------------------------------------------------------------------------
extern "C" void kernel_launch(void* const* d_in, const int* in_sizes, int n_in,
                              void* d_out, int out_size, void* d_ws, size_t ws_size,
                              hipStream_t stream) {
    const float* h      = (const float*)d_in[0];
    const float* mask   = (const float*)d_in[1];
    const int*   idx    = (const int*)d_in[2];
    const float* protos = (const float*)d_in[3];
    float* out = (float*)d_out;
    int n_obs = in_sizes[2];               // 1024

    // workspace layout (floats)
    float* ws_f   = (float*)d_ws;
    float* s      = ws_f;                              // B*N*D   = 524288
    float* alpha  = s + (size_t)BB * NN * DD;          // B*N*K   = 262144
    float* hnum   = alpha + (size_t)BB * NN * KK;      // B*K*J   = 524288
    float* rdenom = hnum + (size_t)BB * KK * JJ;       // B*K     = 128
    int*   flags  = (int*)(rdenom + BB * KK);          // N ints

    flags_zero_kernel<<<(NN + 255) / 256, 256, 0, stream>>>(flags);
    flags_set_kernel<<<(n_obs + 255) / 256, 256, 0, stream>>>(idx, flags, n_obs);

    summary_kernel<<<BB * NN, 64, 0, stream>>>(h, mask, s);
    alpha_kernel<<<BB * NN / 8, 256, 0, stream>>>(s, protos, alpha);

    // B * (JJ/16) waves, 8 waves per block
    hnum_kernel<<<BB * (JJ / 16) / 8, 256, 0, stream>>>(h, alpha, idx, hnum, n_obs);
    rdenom_kernel<<<1, 128, 0, stream>>>(alpha, idx, rdenom, n_obs);

    // B * (N/16) * (JJ/16) waves, 8 waves per block
    impute_kernel<<<BB * (NN / 16) * (JJ / 16) / 8, 256, 0, stream>>>(
        h, alpha, hnum, rdenom, flags, out);
}